// pLayer_64201171140792
// MI455X (gfx1250) — compile-verified
//
#include <hip/hip_runtime.h>
#include <math.h>

typedef float v2f __attribute__((ext_vector_type(2)));
typedef float v8f __attribute__((ext_vector_type(8)));

#define NV   8
#define NE   32768
#define NIN  128
#define NOUT 128
#define NM   130
#define EB   32   // blocks along E per (n-half, v)
#define RT   8    // row-tiles (128 rows) per block: NE/(EB*128)

// ---- workspace layout (floats) ----
static constexpr int O_WP    = 0;        // [V][128][128]
static constexpr int O_WN    = 131072;
static constexpr int O_GA    = 262144;
static constexpr int O_GB    = 393216;
static constexpr int O_BIAS  = 524288;   // [V][128]
static constexpr int O_GP128 = 525312;   // [V][128]
static constexpr int O_GQ128 = 526336;   // [V][128]
static constexpr int O_GCOL  = 527360;   // [128]
static constexpr int O_SX2   = 527488;   // [V][128]  (accumulators start here)
static constexpr int O_SXN2  = 528512;
static constexpr int O_SZ    = 529536;
static constexpr int O_SY2   = 530560;
static constexpr int O_SUZ   = 531584;   // [V]
static constexpr int ACC_LEN = 4 * NV * NIN + NV;   // zeroed each call

__device__ __forceinline__ float t_tanh(float x) {
#if __has_builtin(__builtin_amdgcn_tanhf)
  return __builtin_amdgcn_tanhf(x);
#else
  return tanhf(x);
#endif
}
__device__ __forceinline__ float inv_act(float a) {           // eta_inv = (-0.05,-0.95,0,2)
  return -0.05f - 0.95f * t_tanh(2.0f * a);
}
__device__ __forceinline__ float out_act(float z) {           // eta = (0.01,0.5,0.05,10)
  return 0.01f + 0.5f * t_tanh((z - 0.05f) * 10.0f);
}

// ---------------- init: zero accumulators ----------------
__global__ void init_kernel(float* ws) {
  int i = blockIdx.x * 256 + threadIdx.x;
  if (i < ACC_LEN) ws[O_SX2 + i] = 0.0f;
}

// ---------------- prep: weights / masks / scales ----------------
__global__ __launch_bounds__(128) void prep_kernel(const float* __restrict__ theta_,
                                                   const float* __restrict__ noise_w,
                                                   const float* __restrict__ noise_pos,
                                                   const float* __restrict__ noise_pow,
                                                   float* __restrict__ ws) {
  const int v = blockIdx.x;
  const int n = threadIdx.x;
  const float c1 = inv_act(1.0f);
  // pass 1: column norm G (over all 130 rows) and min|theta_| per column
  float G = 0.0f, g0min = 3.4e38f;
  for (int m = 0; m < NM; m++) {
    float tr = theta_[m * NOUT + n];
    float tc = fminf(fmaxf(tr, -1.0f), 1.0f);
    float th = (fabsf(tc) < 0.01f) ? 0.0f : tc;
    G += fabsf(th * noise_w[(v * NM + m) * NOUT + n]);
    g0min = fminf(g0min, fabsf(tr));
  }
  const float ginv = 0.01f / g0min;   // PGMIN / min_m g0
  float gcol = 0.0f;
  for (int m = 0; m < NM; m++) {
    float tr = theta_[m * NOUT + n];
    float tc = fminf(fmaxf(tr, -1.0f), 1.0f);
    float th = (fabsf(tc) < 0.01f) ? 0.0f : tc;
    float tn = th * noise_w[(v * NM + m) * NOUT + n];
    float W  = fabsf(tn) / (G + 1e-10f);
    float pos = (th * noise_pos[(v * NM + m) * NOUT + n] >= 0.0f) ? 1.0f : 0.0f;
    float p   = (th * noise_pow[(v * NM + m) * NOUT + n] >= 0.0f) ? 1.0f : 0.0f;
    float g   = fabsf(tr) * ginv;
    gcol += g;
    if (m < NIN) {
      int idx = (v * NIN + m) * NOUT + n;
      ws[O_WP + idx] = W * pos;
      ws[O_WN + idx] = W * (1.0f - pos);
      ws[O_GA + idx] = g * p;
      ws[O_GB + idx] = g * (1.0f - p);
    } else if (m == NIN) {  // ones column -> bias; power row-128 terms
      ws[O_BIAS  + v * NOUT + n] = W * pos + c1 * W * (1.0f - pos);
      ws[O_GP128 + v * NOUT + n] = g * p;
      ws[O_GQ128 + v * NOUT + n] = g * (1.0f - p);
    }
  }
  if (v == 0) ws[O_GCOL + n] = gcol;   // g is v-independent
}

// ---------------- stats: Sx2[v,m], Sxn2[v,m] ----------------
__global__ __launch_bounds__(256) void stats_kernel(const float* __restrict__ a,
                                                    float* __restrict__ ws) {
  const int v = blockIdx.y;
  const float* av = a + (size_t)v * NE * NIN;
  const int col = threadIdx.x & (NIN - 1);   // stride keeps column fixed
  float sx2 = 0.0f, sxn2 = 0.0f;
  const int total = NE * NIN;
  for (int i = blockIdx.x * 256 + threadIdx.x; i < total; i += gridDim.x * 256) {
    float x  = av[i];
    float xn = inv_act(x);
    sx2  += x * x;
    sxn2 += xn * xn;
  }
  atomicAdd(&ws[O_SX2  + v * NIN + col], sx2);
  atomicAdd(&ws[O_SXN2 + v * NIN + col], sxn2);
}

// ---------------- main: z, a_new, u.z, Sz, Sy2 via WMMA f32 ----------------
__global__ __launch_bounds__(256) void main_kernel(const float* __restrict__ a,
                                                   float* __restrict__ out,
                                                   float* __restrict__ ws) {
  const int v    = blockIdx.z;
  const int nh   = blockIdx.y;           // n half: 0 or 1 (64 cols each)
  const int bx   = blockIdx.x;
  const int tid  = threadIdx.x;
  const int lane = tid & 31;
  const int wave = tid >> 5;

  __shared__ float sFrag[32768];         // 128 KB: pre-swizzled B fragments (Wp,Wn,Ga,Gb)
  __shared__ float sBias[64];
  __shared__ float sSz[64];
  __shared__ float sSy2[64];
  __shared__ float sSuz;

  // stage: f bits = [kk:5][t:2][mat:2][lane:5][j:1]
  for (int f = tid; f < 32768; f += 256) {
    int j   = f & 1;
    int ln  = (f >> 1) & 31;
    int mat = (f >> 6) & 3;
    int t   = (f >> 8) & 3;
    int kk  = f >> 10;
    int m    = kk * 4 + ((ln >> 4) << 1) + j;
    int ncol = nh * 64 + t * 16 + (ln & 15);
    sFrag[f] = ws[mat * 131072 + (v * NIN + m) * NOUT + ncol];
  }
  if (tid < 64) {
    sBias[tid] = ws[O_BIAS + v * NOUT + nh * 64 + tid];
    sSz[tid] = 0.0f;
    sSy2[tid] = 0.0f;
  }
  if (tid == 0) sSuz = 0.0f;
  __syncthreads();

  const float* aBase = a   + (size_t)v * NE * NIN;
  float*       oBase = out + (size_t)v * NE * NIN;
  float suz = 0.0f;

  for (int rt = 0; rt < RT; rt++) {
    const int rowbase = (bx * RT + rt) * 128 + wave * 16;
    const float* ap = aBase + (size_t)(rowbase + (lane & 15)) * NIN + ((lane >> 4) << 1);

    v8f accz[4] = {};
    v8f accu[4] = {};

#pragma unroll 4
    for (int kk = 0; kk < 32; kk++) {
      v2f ax = *(const v2f*)(ap + kk * 4);
      v2f axn;
      axn.x = inv_act(ax.x);
      axn.y = inv_act(ax.y);
      const float* fb = sFrag + kk * 1024;
#pragma unroll
      for (int t = 0; t < 4; t++) {
        v2f bwp = *(const v2f*)(fb + (t * 4 + 0) * 64 + lane * 2);
        v2f bwn = *(const v2f*)(fb + (t * 4 + 1) * 64 + lane * 2);
        v2f bga = *(const v2f*)(fb + (t * 4 + 2) * 64 + lane * 2);
        v2f bgb = *(const v2f*)(fb + (t * 4 + 3) * 64 + lane * 2);
        accz[t] = __builtin_amdgcn_wmma_f32_16x16x4_f32(false, ax,  false, bwp, (short)0, accz[t], false, false);
        accz[t] = __builtin_amdgcn_wmma_f32_16x16x4_f32(false, axn, false, bwn, (short)0, accz[t], false, false);
        accu[t] = __builtin_amdgcn_wmma_f32_16x16x4_f32(false, ax,  false, bga, (short)0, accu[t], false, false);
        accu[t] = __builtin_amdgcn_wmma_f32_16x16x4_f32(false, axn, false, bgb, (short)0, accu[t], false, false);
      }
    }

    // epilogue: bias, activation store, stats
    const int nl = lane & 15;
    const int hh = lane >> 4;
#pragma unroll
    for (int t = 0; t < 4; t++) {
      float bias = sBias[t * 16 + nl];
      int   ng   = nh * 64 + t * 16 + nl;
      float szl = 0.0f, sy2l = 0.0f;
#pragma unroll
      for (int i = 0; i < 8; i++) {
        float z = accz[t][i] + bias;
        suz  += accu[t][i] * z;
        szl  += z;
        sy2l += z * z;
        int e = rowbase + i + hh * 8;
        oBase[(size_t)e * NOUT + ng] = out_act(z);
      }
      szl  += __shfl_xor(szl, 16, 32);
      sy2l += __shfl_xor(sy2l, 16, 32);
      if (lane < 16) {
        atomicAdd(&sSz[t * 16 + nl], szl);
        atomicAdd(&sSy2[t * 16 + nl], sy2l);
      }
    }
  }

  // wave-reduce u.z, then block accumulate
  for (int off = 16; off >= 1; off >>= 1) suz += __shfl_xor(suz, off, 32);
  if (lane == 0) atomicAdd(&sSuz, suz);
  __syncthreads();
  if (tid < 64) {
    atomicAdd(&ws[O_SZ  + v * NOUT + nh * 64 + tid], sSz[tid]);
    atomicAdd(&ws[O_SY2 + v * NOUT + nh * 64 + tid], sSy2[tid]);
  }
  if (tid == 0) atomicAdd(&ws[O_SUZ + v], sSuz);
}

// ---------------- finalize: power scalar ----------------
__global__ __launch_bounds__(256) void finalize_kernel(const float* __restrict__ ws,
                                                       float* __restrict__ out_power) {
  __shared__ float red[256];
  const int tid = threadIdx.x;
  const float c1 = inv_act(1.0f);
  float acc = 0.0f;
  // T1: sum_{v,m<128} Sx2*sum_n Ga + Sxn2*sum_n Gb
  for (int idx = tid; idx < NV * NIN * NOUT; idx += 256) {
    int v = idx >> 14;
    int m = (idx >> 7) & 127;
    acc += ws[O_GA + idx] * ws[O_SX2  + v * NIN + m]
         + ws[O_GB + idx] * ws[O_SXN2 + v * NIN + m];
  }
  // T2 (m=128 Sx2=E / Sxn2=E*c1^2), T4 (-2*coef*Sz), T5 (Gcol*Sy2)
  for (int idx = tid; idx < NV * NOUT; idx += 256) {
    int n = idx & 127;
    float gp = ws[O_GP128 + idx];
    float gq = ws[O_GQ128 + idx];
    acc += (float)NE * (gp + c1 * c1 * gq);
    acc += -2.0f * (gp + c1 * gq) * ws[O_SZ + idx];
    acc += ws[O_GCOL + n] * ws[O_SY2 + idx];
  }
  // T3: -2 * sum_v u.z
  if (tid < NV) acc += -2.0f * ws[O_SUZ + tid];
  red[tid] = acc;
  __syncthreads();
  for (int s = 128; s > 0; s >>= 1) {
    if (tid < s) red[tid] += red[tid + s];
    __syncthreads();
  }
  if (tid == 0) *out_power = red[0] / (float)((size_t)NE * NV);
}

extern "C" void kernel_launch(void* const* d_in, const int* in_sizes, int n_in,
                              void* d_out, int out_size, void* d_ws, size_t ws_size,
                              hipStream_t stream) {
  const float* a      = (const float*)d_in[0];
  const float* theta_ = (const float*)d_in[1];
  const float* nw     = (const float*)d_in[2];
  const float* npos   = (const float*)d_in[3];
  const float* npow   = (const float*)d_in[4];
  float* out = (float*)d_out;
  float* ws  = (float*)d_ws;

  init_kernel<<<(ACC_LEN + 255) / 256, 256, 0, stream>>>(ws);
  prep_kernel<<<NV, 128, 0, stream>>>(theta_, nw, npos, npow, ws);
  stats_kernel<<<dim3(64, NV), 256, 0, stream>>>(a, ws);
  main_kernel<<<dim3(EB, 2, NV), 256, 0, stream>>>(a, out, ws);
  finalize_kernel<<<1, 256, 0, stream>>>(ws, out + (size_t)NV * NE * NIN);
}